// AdaptiveMedianBaseline_41893111005597
// MI455X (gfx1250) — compile-verified
//
#include <hip/hip_runtime.h>
#include <hip/hip_bf16.h>
#include <math.h>

// Problem constants from the reference
#define Bc   2
#define Cc   1
#define Tc   8640
#define Fc   8
#define HIDc 16

#define WAVES_PER_WG 8
#define OUTS_PER_WG  32

typedef __attribute__((ext_vector_type(2))) float v2f;
typedef __attribute__((ext_vector_type(8))) float v8f;

// Order-preserving float -> uint key (total order matches float <)
__device__ __forceinline__ unsigned f2key(float f) {
    unsigned u = __float_as_uint(f);
    return u ^ ((u & 0x80000000u) ? 0xFFFFFFFFu : 0x80000000u);
}
__device__ __forceinline__ float key2f(unsigned k) {
    unsigned u = (k & 0x80000000u) ? (k ^ 0x80000000u) : ~k;
    return __uint_as_float(u);
}

// ---------------------------------------------------------------------------
// Kernel 1: MLP gate via V_WMMA_F32_16X16X4_F32 (one wave).
//   H = relu(meta @ w1 + b1); factor = sigmoid(H @ w2 + b2)
//   w = min(max_b int(hours_b * 360), T); derive pad/midx/n_out.
// A tile (16x4 f32): lanes 0-15 hold rows M; VGPR0 = K{0|2}, VGPR1 = K{1|3}
// (K-half selected by lane>=16). B tile (4x16) mirrors the half striping.
// D tile rows 0/1 -> c[0], c[1] on lanes 0..15 (N = lane).
// ---------------------------------------------------------------------------
__global__ void AdaptiveMedian_mlp_kernel(const float* __restrict__ meta,
                                          const float* __restrict__ w1,
                                          const float* __restrict__ b1,
                                          const float* __restrict__ w2,
                                          const float* __restrict__ b2,
                                          int* __restrict__ params) {
    const int lane = threadIdx.x & 31;
    const int m    = lane & 15;      // A-tile row
    const int half = lane >> 4;      // K-half selector
    const int n    = lane & 15;      // B/C/D column

    const float rowScale = (m < Bc) ? 1.0f : 0.0f;  // zero-pad rows 2..15
    const int   mm  = (m < Bc) ? m : 0;             // keep loads in-bounds
    const int   k0  = 2 * half;                     // K within a 4-chunk
    const int   k1  = 2 * half + 1;

    v2f a0, a1, bt0, bt1;
    a0.x = meta[mm * Fc + k0]       * rowScale;     // K = 0..3
    a0.y = meta[mm * Fc + k1]       * rowScale;
    a1.x = meta[mm * Fc + 4 + k0]   * rowScale;     // K = 4..7
    a1.y = meta[mm * Fc + 4 + k1]   * rowScale;
    bt0.x = w1[(k0)     * HIDc + n];
    bt0.y = w1[(k1)     * HIDc + n];
    bt1.x = w1[(4 + k0) * HIDc + n];
    bt1.y = w1[(4 + k1) * HIDc + n];

    v8f c = {};
    c = __builtin_amdgcn_wmma_f32_16x16x4_f32(false, a0, false, bt0,
                                              (short)0, c, false, false);
    c = __builtin_amdgcn_wmma_f32_16x16x4_f32(false, a1, false, bt1,
                                              (short)0, c, false, false);

    // Second layer: relu + weighted lane values, butterfly-reduce lanes 0..15
    float h0 = fmaxf(c[0] + b1[n], 0.0f) * w2[n];
    float h1 = fmaxf(c[1] + b1[n], 0.0f) * w2[n];
    #pragma unroll
    for (int mvs = 8; mvs >= 1; mvs >>= 1) {
        h0 += __shfl_xor(h0, mvs, 32);
        h1 += __shfl_xor(h1, mvs, 32);
    }

    if (lane == 0) {
        const float bb = b2[0];
        const float f0 = 1.0f / (1.0f + __expf(-(h0 + bb)));
        const float f1 = 1.0f / (1.0f + __expf(-(h1 + bb)));
        const int s0 = (int)((4.0f + f0 * 44.0f) * 360.0f);  // trunc == astype(i32)
        const int s1 = (int)((4.0f + f1 * 44.0f) * 360.0f);
        int w = s0 > s1 ? s0 : s1;
        if (w > Tc) w = Tc;
        const int pad   = w >> 1;
        const int midx  = (w - 1) >> 1;
        const int n_out = Tc + 2 * pad - w + 1;
        params[0] = w; params[1] = pad; params[2] = midx; params[3] = n_out;
    }
}

// ---------------------------------------------------------------------------
// Kernel 2: sliding lower-median via wave-cooperative 4-bit MSB radix select.
// One row of keys resident in LDS per workgroup; one wave per output position;
// edge padding handled by index clamping.
// ---------------------------------------------------------------------------
__global__ void AdaptiveMedian_select_kernel(const float* __restrict__ att,
                                             const int* __restrict__ params,
                                             float* __restrict__ out) {
    __shared__ unsigned s_keys[Tc];

    const int row  = blockIdx.x;           // 0..B*C-1
    const int tid  = threadIdx.x;
    const int lane = tid & 31;
    const int wave = tid >> 5;

    const int w     = params[0];
    const int pad   = params[1];
    const int midx  = params[2];
    const int n_out = params[3];

    const float* rowp = att + (size_t)row * Tc;
    for (int i = tid; i < Tc; i += blockDim.x) s_keys[i] = f2key(rowp[i]);
    __syncthreads();

    const int base = blockIdx.y * OUTS_PER_WG;
    for (int o = base + wave; o < base + OUTS_PER_WG; o += WAVES_PER_WG) {
        if (o >= n_out) continue;

        unsigned pfx = 0u, gmask = 0u;     // fixed high bits / their mask
        unsigned rank = (unsigned)midx;    // k-th smallest within group
        const int start = o - pad;         // window source = clamp(start + j)

        #pragma unroll
        for (int shift = 28; shift >= 0; shift -= 4) {
            unsigned cnt[16];
            #pragma unroll
            for (int b = 0; b < 16; ++b) cnt[b] = 0u;

            for (int j = lane; j < w; j += 32) {
                int src = start + j;
                src = src < 0 ? 0 : (src > Tc - 1 ? Tc - 1 : src);
                const unsigned key = s_keys[src];
                const unsigned dd =
                    (((key ^ pfx) & gmask) == 0u) ? ((key >> shift) & 0xFu) : 16u;
                #pragma unroll
                for (int b = 0; b < 16; ++b) cnt[b] += (dd <= (unsigned)b) ? 1u : 0u;
            }

            // Wave-wide butterfly reduction of the 16 cumulative counters
            #pragma unroll
            for (int b = 0; b < 16; ++b) {
                unsigned v = cnt[b];
                #pragma unroll
                for (int mvs = 16; mvs >= 1; mvs >>= 1)
                    v += (unsigned)__shfl_xor((int)v, mvs, 32);
                cnt[b] = v;
            }

            // Smallest digit D with cumulative count > rank (branchless)
            unsigned D = 0u, below = 0u;
            #pragma unroll
            for (int b = 0; b < 15; ++b) {
                const bool adv = (cnt[b] <= rank);
                D += adv ? 1u : 0u;
                below = adv ? cnt[b] : below;
            }
            rank -= below;
            pfx  |= D << shift;
            gmask |= 0xFu << shift;
        }

        if (lane == 0) out[(size_t)row * n_out + o] = key2f(pfx);
    }
}

extern "C" void kernel_launch(void* const* d_in, const int* in_sizes, int n_in,
                              void* d_out, int out_size, void* d_ws, size_t ws_size,
                              hipStream_t stream) {
    const float* att  = (const float*)d_in[0];
    const float* meta = (const float*)d_in[1];
    const float* w1   = (const float*)d_in[2];
    const float* b1   = (const float*)d_in[3];
    const float* w2   = (const float*)d_in[4];
    const float* b2   = (const float*)d_in[5];
    float* out   = (float*)d_out;
    int* params  = (int*)d_ws;

    AdaptiveMedian_mlp_kernel<<<1, 32, 0, stream>>>(meta, w1, b1, w2, b2, params);

    const int rows   = Bc * Cc;
    const int maxOut = Tc + 1;  // n_out is T or T+1; guard inside kernel
    dim3 grid(rows, (maxOut + OUTS_PER_WG - 1) / OUTS_PER_WG);
    AdaptiveMedian_select_kernel<<<grid, WAVES_PER_WG * 32, 0, stream>>>(att, params, out);
}